// LinearAttention2_63118839382215
// MI455X (gfx1250) — compile-verified
//
#include <hip/hip_runtime.h>
#include <hip/hip_bf16.h>

typedef __attribute__((ext_vector_type(2))) float v2f;
typedef __attribute__((ext_vector_type(8))) float v8f;

#define NB   4
#define LSEQ 16384
#define HH   8
#define DD   64
#define NH   (NB*HH)          // 32
#define CHUNKS 32             // seq chunks per (n,h)
#define ROWS_PER_BLOCK (LSEQ/CHUNKS)   // 512
#define SUB  64               // rows staged per LDS iteration
#define NSUB (ROWS_PER_BLOCK/SUB)      // 8
#define ROWSTRIDE (HH*DD)     // 512 floats between seq rows

__device__ __forceinline__ float elu1(float x) {
    return x > 0.0f ? x + 1.0f : __expf(x);
}

// ---------------- Phase 1: KV[m,d] = sum_s fk(K[s,d]) * V[s,m]; ksum[d] = sum_s fk(K[s,d])
// One block (64 thr / 2 waves) handles one (n,h, 512-row chunk). Each wave builds a full
// 64x64 fp32 accumulator as 4x4 tiles of v_wmma_f32_16x16x4_f32 (inner dim = sequence).
__global__ void __launch_bounds__(64) la_phase1(const float* __restrict__ Kin,
                                                const float* __restrict__ Vin,
                                                float* __restrict__ kvPart,
                                                float* __restrict__ ksPart) {
    __shared__ float kbuf[SUB * DD];   // 16 KB, elu-applied K rows; reused as reduction buf (wave0)
    __shared__ float vbuf[SUB * DD];   // 16 KB, raw V rows;        reused as reduction buf (wave1)

    const int tid  = threadIdx.x;
    const int lane = tid & 31;
    const int wave = tid >> 5;
    const int half = lane >> 4;     // which 16-lane half
    const int lm   = lane & 15;

    const int blk   = blockIdx.x;
    const int nh    = blk / CHUNKS;
    const int chunk = blk % CHUNKS;
    const int n = nh / HH, h = nh % HH;

    const long base = ((long)n * LSEQ) * ROWSTRIDE + (long)h * DD;
    const int  s0   = chunk * ROWS_PER_BLOCK;

    v8f acc[4][4] = {};
    float kcol = 0.0f;              // running column sum for d = tid (ksum partial)

    for (int sub = 0; sub < NSUB; ++sub) {
        const int sbase = s0 + sub * SUB;
        __syncthreads();            // previous sub-chunk fully consumed before overwrite
        // ---- coalesced staging: iteration i loads row (sbase+i), col = tid
        for (int i = 0; i < SUB; ++i) {
            const long off = base + (long)(sbase + i) * ROWSTRIDE + tid;
            float kv = Kin[off];
            kv = elu1(kv);
            kbuf[i * DD + tid] = kv;
            kcol += kv;
            vbuf[i * DD + tid] = Vin[off];
        }
        if (sub + 1 < NSUB) {       // prefetch next sub-chunk (global_prefetch_b8)
            const long poff = base + (long)(sbase + SUB) * ROWSTRIDE + tid;
            __builtin_prefetch(&Kin[poff], 0, 3);
            __builtin_prefetch(&Vin[poff], 0, 3);
        }
        __syncthreads();
        // ---- WMMA accumulation: wave w consumes staged rows [w*32, w*32+32) => 8 K-steps of 4
        const int rbase = wave * 32;
        for (int ks = 0; ks < 8; ++ks) {
            const int r0 = rbase + ks * 4 + half * 2;   // this half-wave's two K rows
            v2f a[4], b[4];
            #pragma unroll
            for (int mt = 0; mt < 4; ++mt) {            // A = V fragment (16x4, M=m, K=s)
                a[mt].x = vbuf[r0 * DD + mt * 16 + lm];
                a[mt].y = vbuf[(r0 + 1) * DD + mt * 16 + lm];
            }
            #pragma unroll
            for (int dt = 0; dt < 4; ++dt) {            // B = K fragment (4x16, K=s, N=d)
                b[dt].x = kbuf[r0 * DD + dt * 16 + lm];
                b[dt].y = kbuf[(r0 + 1) * DD + dt * 16 + lm];
            }
            #pragma unroll
            for (int mt = 0; mt < 4; ++mt)
                #pragma unroll
                for (int dt = 0; dt < 4; ++dt)
                    acc[mt][dt] = __builtin_amdgcn_wmma_f32_16x16x4_f32(
                        false, a[mt], false, b[dt], (short)0, acc[mt][dt], false, false);
        }
    }

    // ---- deterministic cross-wave reduction through LDS (reuse staging buffers)
    __syncthreads();
    float* red = (wave == 0) ? kbuf : vbuf;
    #pragma unroll
    for (int mt = 0; mt < 4; ++mt)
        #pragma unroll
        for (int dt = 0; dt < 4; ++dt)
            #pragma unroll
            for (int r = 0; r < 8; ++r) {
                const int m  = mt * 16 + r + 8 * half;  // C/D layout: M = r (+8 for hi half)
                const int dv = dt * 16 + lm;            //             N = lane%16
                red[m * DD + dv] = acc[mt][dt][r];
            }
    __syncthreads();
    float* __restrict__ outp = kvPart + (long)blk * 4096;
    for (int i = tid; i < 4096; i += 64)
        outp[i] = kbuf[i] + vbuf[i];
    ksPart[blk * 64 + tid] = kcol;
}

// ---------------- Phase 1b: deterministic reduction of chunk partials per (n,h)
__global__ void __launch_bounds__(256) la_reduce(const float* __restrict__ kvPart,
                                                 const float* __restrict__ ksPart,
                                                 float* __restrict__ KVf,
                                                 float* __restrict__ KSf) {
    const int nh = blockIdx.x;
    const int tid = threadIdx.x;
    for (int loc = tid; loc < 4096; loc += 256) {
        float s = 0.0f;
        for (int c = 0; c < CHUNKS; ++c)
            s += kvPart[(long)(nh * CHUNKS + c) * 4096 + loc];
        KVf[nh * 4096 + loc] = s;
    }
    if (tid < 64) {
        float s = 0.0f;
        for (int c = 0; c < CHUNKS; ++c)
            s += ksPart[(nh * CHUNKS + c) * 64 + tid];
        KSf[nh * 64 + tid] = s;
    }
}

// ---------------- Phase 2: out[n,h,l,m] = z[l] * sum_d fq(Q[l,d]) * KV[m,d]
// One block (64 thr / 2 waves) handles 32 l-rows of one (n,h); wave = 16 rows.
__global__ void __launch_bounds__(64) la_phase2(const float* __restrict__ Qin,
                                                const float* __restrict__ KVf,
                                                const float* __restrict__ KSf,
                                                float* __restrict__ Out) {
    __shared__ float kvlds[4096];     // 16 KB: KV[m*64+d]
    __shared__ float qbuf[32 * DD];   // 8 KB: elu(Q) rows
    __shared__ float kslds[DD];
    __shared__ float zlds[32];

    const int tid  = threadIdx.x;
    const int lane = tid & 31;
    const int wave = tid >> 5;
    const int half = lane >> 4;
    const int lm   = lane & 15;

    const int blk    = blockIdx.x;
    const int nh     = blk >> 9;          // / (L/32 = 512)
    const int lchunk = blk & 511;
    const int n = nh / HH, h = nh % HH;
    const int l0 = lchunk * 32;
    const long qbase = ((long)n * LSEQ) * ROWSTRIDE + (long)h * DD;

    for (int i = tid; i < 4096; i += 64) kvlds[i] = KVf[nh * 4096 + i];
    kslds[tid] = KSf[nh * 64 + tid];
    for (int i = 0; i < 32; ++i) {
        const float qv = Qin[qbase + (long)(l0 + i) * ROWSTRIDE + tid];
        qbuf[i * DD + tid] = elu1(qv);
    }
    __syncthreads();

    if (tid < 32) {                        // z[l] = 1/(q . ksum + eps)
        float dot = 0.0f;
        for (int dd = 0; dd < DD; ++dd) dot += qbuf[tid * DD + dd] * kslds[dd];
        zlds[tid] = 1.0f / (dot + 1e-6f);
    }
    __syncthreads();

    const int rrow = wave * 16;            // this wave's l-row tile base (within block)
    v8f acc[4] = {};
    for (int ds = 0; ds < 16; ++ds) {      // inner dim d in steps of 4
        const int d0 = ds * 4 + half * 2;
        v2f a;                             // A = Q fragment (16x4, M=l, K=d)
        a.x = qbuf[(rrow + lm) * DD + d0];
        a.y = qbuf[(rrow + lm) * DD + d0 + 1];
        #pragma unroll
        for (int mt = 0; mt < 4; ++mt) {   // B = KV^T fragment (4x16, K=d, N=m)
            v2f b;
            b.x = kvlds[(mt * 16 + lm) * DD + d0];
            b.y = kvlds[(mt * 16 + lm) * DD + d0 + 1];
            acc[mt] = __builtin_amdgcn_wmma_f32_16x16x4_f32(
                false, a, false, b, (short)0, acc[mt], false, false);
        }
    }

    // out index: ((n*H + h)*L + l)*64 + m   (reference's final swapaxes(1,2))
    const long obase = ((long)(n * HH + h) * LSEQ + (l0 + rrow)) * 64;
    #pragma unroll
    for (int r = 0; r < 8; ++r) {
        const int lrow = r + 8 * half;
        const float z = zlds[rrow + lrow];
        #pragma unroll
        for (int mt = 0; mt < 4; ++mt)
            Out[obase + (long)lrow * 64 + mt * 16 + lm] = acc[mt][r] * z;
    }
}

extern "C" void kernel_launch(void* const* d_in, const int* in_sizes, int n_in,
                              void* d_out, int out_size, void* d_ws, size_t ws_size,
                              hipStream_t stream) {
    const float* Q = (const float*)d_in[0];
    const float* K = (const float*)d_in[1];
    const float* V = (const float*)d_in[2];
    float* out = (float*)d_out;

    float* ws     = (float*)d_ws;
    float* kvPart = ws;                                   // 1024 * 4096
    float* ksPart = kvPart + (size_t)NH * CHUNKS * 4096;  // 1024 * 64
    float* KVf    = ksPart + (size_t)NH * CHUNKS * 64;    // 32 * 4096
    float* KSf    = KVf    + (size_t)NH * 4096;           // 32 * 64

    la_phase1<<<NH * CHUNKS, 64, 0, stream>>>(K, V, kvPart, ksPart);
    la_reduce<<<NH, 256, 0, stream>>>(kvPart, ksPart, KVf, KSf);
    la_phase2<<<NH * (LSEQ / 32), 64, 0, stream>>>(Q, KVf, KSf, out);
}